// MyModel_11879879543894
// MI455X (gfx1250) — compile-verified
//
#include <hip/hip_runtime.h>

// ---------------------------------------------------------------------------
// GraphSAGE (2 layers, mean aggregation) for MI455X / gfx1250.
//   h   = mean_scatter(x)   @ W_l1 + b_l1 + x @ W_r1     [50000 x 256]
//   out = mean_scatter(h)   @ W_l2 + b_l2 + h @ W_r2     [50000 x 128]
// Scatter/gather is L2-resident atomic traffic (dominant cost);
// all four GEMMs run on V_WMMA_F32_16X16X4_F32 (exact f32).
// ---------------------------------------------------------------------------

#define N_NODES 50000
#define D_IN    128
#define D_HID   256
#define D_OUT   128

typedef __attribute__((ext_vector_type(2))) float v2f;
typedef __attribute__((ext_vector_type(8))) float v8f;

// ---------------- utility kernels ----------------

__global__ void zero_f32(float* __restrict__ p, long n) {
  long i = (long)blockIdx.x * blockDim.x + threadIdx.x;
  long stride = (long)gridDim.x * blockDim.x;
  for (; i < n; i += stride) p[i] = 0.0f;
}

__global__ void degree_count(const int* __restrict__ dst, int* __restrict__ deg, int E) {
  int e = blockIdx.x * blockDim.x + threadIdx.x;
  if (e < E) atomicAdd(&deg[dst[e]], 1);
}

__global__ void inv_degree(const int* __restrict__ deg, float* __restrict__ inv, int N) {
  int i = blockIdx.x * blockDim.x + threadIdx.x;
  if (i < N) inv[i] = 1.0f / fmaxf((float)deg[i], 1.0f);
}

// ---------------- edge scatter-add (message aggregation) ----------------
// D/4 lanes cooperate per edge; each lane moves one float4 via L2 atomics.

template <int D>
__global__ void scatter_add(const float* __restrict__ feat,
                            const int* __restrict__ src,
                            const int* __restrict__ dst,
                            float* __restrict__ acc, int E) {
  constexpr int LPE = D / 4;                 // lanes per edge
  int gid = blockIdx.x * blockDim.x + threadIdx.x;
  int e = gid / LPE;
  if (e >= E) return;
  int c = (gid % LPE) * 4;
  int s = src[e];
  int d = dst[e];
  const float4 v = *(const float4*)(feat + (size_t)s * D + c);
  float* p = acc + (size_t)d * D + c;
  atomicAdd(p + 0, v.x);
  atomicAdd(p + 1, v.y);
  atomicAdd(p + 2, v.z);
  atomicAdd(p + 3, v.w);
}

// ---------------- fused SAGE linear: out = (inv.*acc)@Wl + b + X@Wr -------
// One wave computes a 16x32 output slab: two 16x16 WMMA accumulators that
// share the A fragment (independent WMMAs per K-step hide the RAW hazard).
// A fragment (f32 16x4, 2 VGPR): lanes 0-15 -> K=k,k+1 ; lanes 16-31 -> K=k+2,k+3.
// B fragment mirrored; C/D: row = r + 8*(lane>=16), col = lane&15.

__global__ void sage_linear_wmma(const float* __restrict__ agg,
                                 const float* __restrict__ invd,
                                 const float* __restrict__ xin,
                                 const float* __restrict__ Wl,
                                 const float* __restrict__ bias,
                                 const float* __restrict__ Wr,
                                 float* __restrict__ out,
                                 int N, int K, int Dout) {
  const int wave = (blockIdx.x * blockDim.x + threadIdx.x) >> 5;  // wave32
  const int lane = threadIdx.x & 31;
  const int colPairs = Dout >> 5;            // 16x32 slabs per row band
  const int rowTile = wave / colPairs;
  const int colPair = wave % colPairs;
  if (rowTile >= (N >> 4)) return;           // wave-uniform: EXEC stays all-1s

  const int row0 = rowTile << 4;
  const int col0 = colPair << 5;
  const int ml   = lane & 15;
  const int hi   = lane >> 4;                // 0 | 1
  const int kh   = hi << 1;                  // 0 | 2 : K sub-offset of this half
  const int m    = row0 + ml;                // A row owned by this lane
  const int n0   = col0 + ml;                // B/D col, tile 0
  const int n1   = n0 + 16;                  // B/D col, tile 1

  v8f c0 = {};
  v8f c1 = {};

  const float s = invd[m];                   // fold 1/max(deg,1) into A load
  const float* arow = agg + (size_t)m * K;
  const float* xrow = xin + (size_t)m * K;

  // pass 1: scaled neighbor mean @ Wl
  for (int k = 0; k < K; k += 4) {
    v2f a = *(const v2f*)(arow + k + kh);
    a *= s;
    const float* wk = Wl + (size_t)(k + kh) * Dout;
    v2f b0, b1;
    b0.x = wk[n0];          b1.x = wk[n1];
    b0.y = wk[Dout + n0];   b1.y = wk[Dout + n1];
    c0 = __builtin_amdgcn_wmma_f32_16x16x4_f32(false, a, false, b0, (short)0, c0, false, false);
    c1 = __builtin_amdgcn_wmma_f32_16x16x4_f32(false, a, false, b1, (short)0, c1, false, false);
  }
  // pass 2: self features @ Wr
  for (int k = 0; k < K; k += 4) {
    v2f a = *(const v2f*)(xrow + k + kh);
    const float* wk = Wr + (size_t)(k + kh) * Dout;
    v2f b0, b1;
    b0.x = wk[n0];          b1.x = wk[n1];
    b0.y = wk[Dout + n0];   b1.y = wk[Dout + n1];
    c0 = __builtin_amdgcn_wmma_f32_16x16x4_f32(false, a, false, b0, (short)0, c0, false, false);
    c1 = __builtin_amdgcn_wmma_f32_16x16x4_f32(false, a, false, b1, (short)0, c1, false, false);
  }

  const float bb0 = bias[n0];
  const float bb1 = bias[n1];
  const int rbase = row0 + (hi << 3);
#pragma unroll
  for (int r = 0; r < 8; ++r) {
    float* orow = out + (size_t)(rbase + r) * Dout;
    orow[n0] = c0[r] + bb0;
    orow[n1] = c1[r] + bb1;
  }
}

// ---------------------------------------------------------------------------

static inline size_t align256(size_t b) { return (b + 255) & ~(size_t)255; }

extern "C" void kernel_launch(void* const* d_in, const int* in_sizes, int n_in,
                              void* d_out, int out_size, void* d_ws, size_t ws_size,
                              hipStream_t stream) {
  const float* x   = (const float*)d_in[0];
  const int*   ei  = (const int*)d_in[1];
  const float* Wl1 = (const float*)d_in[2];
  const float* bl1 = (const float*)d_in[3];
  const float* Wr1 = (const float*)d_in[4];
  const float* Wl2 = (const float*)d_in[5];
  const float* bl2 = (const float*)d_in[6];
  const float* Wr2 = (const float*)d_in[7];
  float* out = (float*)d_out;

  const int N = in_sizes[0] / D_IN;      // 50000
  const int E = in_sizes[1] / 2;         // 800000
  const int* src = ei;
  const int* dst = ei + E;

  // workspace carve-up (all L2-resident):
  //   deg  : N int32
  //   invd : N f32
  //   h    : N x 256 f32 (layer-1 output)
  //   buf  : N x 256 f32 (acc1 uses first half, then reused as acc2)
  char* ws = (char*)d_ws;
  int*   deg  = (int*)ws;                       ws += align256((size_t)N * 4);
  float* invd = (float*)ws;                     ws += align256((size_t)N * 4);
  float* h    = (float*)ws;                     ws += align256((size_t)N * D_HID * 4);
  float* buf  = (float*)ws;
  (void)ws_size; (void)n_in; (void)out_size;

  const int TB = 256;

  // ---- shared degree / inverse-degree (identical for both layers) ----
  zero_f32<<<(N + TB - 1) / TB, TB, 0, stream>>>((float*)deg, (long)N);
  degree_count<<<(E + TB - 1) / TB, TB, 0, stream>>>(dst, deg, E);
  inv_degree<<<(N + TB - 1) / TB, TB, 0, stream>>>(deg, invd, N);

  // ---- layer 1: aggregate x (D=128), fused linear -> h [N,256] ----
  {
    long nacc = (long)N * D_IN;
    zero_f32<<<(int)((nacc + TB - 1) / TB), TB, 0, stream>>>(buf, nacc);
    long nthr = (long)E * (D_IN / 4);
    scatter_add<D_IN><<<(int)((nthr + TB - 1) / TB), TB, 0, stream>>>(x, src, dst, buf, E);
    int waves = (N / 16) * (D_HID / 32);
    int blocks = (waves * 32 + TB - 1) / TB;
    sage_linear_wmma<<<blocks, TB, 0, stream>>>(buf, invd, x, Wl1, bl1, Wr1, h,
                                                N, D_IN, D_HID);
  }

  // ---- layer 2: aggregate h (D=256), fused linear -> out [N,128] ----
  {
    long nacc = (long)N * D_HID;
    zero_f32<<<(int)((nacc + TB - 1) / TB), TB, 0, stream>>>(buf, nacc);
    long nthr = (long)E * (D_HID / 4);
    scatter_add<D_HID><<<(int)((nthr + TB - 1) / TB), TB, 0, stream>>>(h, src, dst, buf, E);
    int waves = (N / 16) * (D_OUT / 32);
    int blocks = (waves * 32 + TB - 1) / TB;
    sage_linear_wmma<<<blocks, TB, 0, stream>>>(buf, invd, h, Wl2, bl2, Wr2, out,
                                                N, D_HID, D_OUT);
  }
}